// Yolo_loss_86053964743131
// MI455X (gfx1250) — compile-verified
//
#include <hip/hip_runtime.h>
#include <hip/hip_bf16.h>
#include <stdint.h>

// ---------------- problem constants (from reference) ----------------
#define B_    128
#define M_    60
#define F_    19
#define A_    3
#define NC_   80
#define NCH_  85          // 5 + NC
#define FF_   (F_ * F_)   // 361
#define NB_   (B_ * A_)   // 384 stage-1 blocks
#define TRUTH_STRIDE 16
#define TRUTH_FLOATS (B_ * M_ * TRUTH_STRIDE)  // 122880 floats in workspace

typedef __attribute__((ext_vector_type(2))) float        v2f;
typedef __attribute__((ext_vector_type(8))) float        v8f;
typedef __attribute__((ext_vector_type(4))) unsigned int u32x4;
typedef __attribute__((ext_vector_type(8))) int          i32x8;
typedef __attribute__((ext_vector_type(4))) int          i32x4;

// anchors / STRIDE, grid units
__device__ __constant__ float c_anch[9][2] = {
    {10.f/32.f, 13.f/32.f}, {16.f/32.f, 30.f/32.f}, {33.f/32.f, 23.f/32.f},
    {30.f/32.f, 61.f/32.f}, {62.f/32.f, 45.f/32.f}, {59.f/32.f,119.f/32.f},
    {116.f/32.f, 90.f/32.f},{156.f/32.f,198.f/32.f},{373.f/32.f,326.f/32.f}};

__device__ __forceinline__ float sigmoidf(float x) { return 1.0f / (1.0f + __expf(-x)); }

// torch BCELoss(sum) elementwise term with -100 clamp
__device__ __forceinline__ float bce_term(float p, float t, float w) {
    float lp = fmaxf(__logf(p), -100.0f);
    float lq = fmaxf(__logf(1.0f - p), -100.0f);
    return -w * (t * lp + (1.0f - t) * lq);
}

// ------------------------------------------------------------------
// Kernel 1: per-truth preprocessing -> 16-float record per (b, m)
// [tx, ty, tw, th, valid, matched, best_n, ti, tj, twt, tht, scale, cls, 0,0,0]
// ------------------------------------------------------------------
__global__ void yolo_truth_kernel(const float* __restrict__ labels,
                                  float* __restrict__ truth) {
    int idx = blockIdx.x * blockDim.x + threadIdx.x;
    if (idx >= B_ * M_) return;
    const float* L = labels + (size_t)idx * 5;
    float x1 = L[0], y1 = L[1], x2 = L[2], y2 = L[3], cls = L[4];
    float valid = ((x1 + y1 + x2 + y2 + cls) > 0.f) ? 1.f : 0.f;
    float tx = floorf(x1 / 32.f), ty = floorf(y1 / 32.f);
    float tw = floorf((x2 - x1) / 32.f), th = floorf((y2 - y1) / 32.f);
    int ti = (int)tx, tj = (int)ty;

    // anchor match: IoU of (0,0,tw,th) vs 9 ref anchors; first max wins (argmax)
    float bestIou = -1.f; int best = 0;
    #pragma unroll
    for (int n = 0; n < 9; ++n) {
        float aw = c_anch[n][0], ah = c_anch[n][1];
        float mw = fminf(tw, aw), mh = fminf(th, ah);
        float en = (mw > 0.f && mh > 0.f) ? 1.f : 0.f;
        float inter = mw * mh * en;
        float iou = inter / (tw * th + aw * ah - inter);
        if (iou > bestIou) { bestIou = iou; best = n; }
    }
    int bestN = best % 3;
    float matched = (valid > 0.f && best >= 6) ? 1.f : 0.f;
    float aw = c_anch[6 + bestN][0], ah = c_anch[6 + bestN][1];
    float twt = __logf(tw / aw + 1e-16f);
    float tht = __logf(th / ah + 1e-16f);
    float scale = sqrtf(2.f - tw * th / (float)FF_);

    float* T = truth + (size_t)idx * TRUTH_STRIDE;
    T[0] = tx;  T[1] = ty;  T[2] = tw;  T[3] = th;
    T[4] = valid; T[5] = matched; T[6] = (float)bestN;
    T[7] = (float)ti; T[8] = (float)tj;
    T[9] = twt; T[10] = tht; T[11] = scale; T[12] = cls;
    T[13] = 0.f; T[14] = 0.f; T[15] = 0.f;
}

// ------------------------------------------------------------------
// Kernel 2: one block per (b, a); fused activation + ignore-mask +
// match + all loss terms in a single pass over output channels.
// Truth table staged into LDS via the Tensor Data Mover.
// ------------------------------------------------------------------
__global__ __launch_bounds__(384) void yolo_cell_kernel(
    const float* __restrict__ output, const float* __restrict__ truth,
    float* __restrict__ partials) {
    __shared__ float sT[M_ * TRUTH_STRIDE];   // 960 dwords = 3840 B
    __shared__ float sRed[12 * 5];

    const int b = blockIdx.x / A_;
    const int a = blockIdx.x % A_;
    const float* gT = truth + (size_t)b * M_ * TRUTH_STRIDE;

#if __has_builtin(__builtin_amdgcn_tensor_load_to_lds) && __has_builtin(__builtin_amdgcn_s_wait_tensorcnt)
    // TDM: DMA the 16x60-dword truth tile global -> LDS (wave 0 issues once;
    // EXEC is ignored by tensor ops). D# per CDNA5 ISA.
    if (threadIdx.x < 32) {
        uint64_t gaddr  = (uint64_t)(uintptr_t)gT;
        uint32_t ldsOff = (uint32_t)(uintptr_t)(&sT[0]); // low 32b of generic = LDS offset
        u32x4 g0;
        g0[0] = 1u;                                       // count=1 (valid descriptor)
        g0[1] = ldsOff;                                   // lds_addr [63:32]
        g0[2] = (uint32_t)(gaddr & 0xffffffffu);          // global_addr [95:64]
        g0[3] = (uint32_t)((gaddr >> 32) & 0x01ffffffu)   // global_addr [120:96]
              | 0x80000000u;                              // type=2 at bits [127:126]
        i32x8 g1;
        g1[0] = 0x00020000;        // workgroup_mask=0, data_size=2 (4 bytes)
        g1[1] = (int)(16u << 16);  // tensor_dim0 = 16 (bits 79:48 -> low16 here)
        g1[2] = (int)(60u << 16);  // tensor_dim0 hi=0, tensor_dim1 = 60
        g1[3] = (int)(16u << 16);  // tensor_dim1 hi=0, tile_dim0 = 16
        g1[4] = 60;                // tile_dim1 = 60, tile_dim2 = 0
        g1[5] = 16;                // tensor_dim0_stride = 16 (low 32 of 48)
        g1[6] = 0;                 // stride hi + tensor_dim1_stride lo
        g1[7] = 0;
        i32x4 z4 = {0, 0, 0, 0};                  // 2-D tensor: groups 2/3 unused
        i32x8 z8 = {0, 0, 0, 0, 0, 0, 0, 0};      // extra group (6-arg toolchain form)
        __builtin_amdgcn_tensor_load_to_lds(g0, g1, z4, z4, z8, 0);
        __builtin_amdgcn_s_wait_tensorcnt(0);
    }
#else
    for (int k = threadIdx.x; k < M_ * TRUTH_STRIDE; k += blockDim.x) sT[k] = gT[k];
#endif
    __syncthreads();

    float lxy = 0.f, lwh = 0.f, lobj = 0.f, lcls = 0.f, ll2 = 0.f;
    const int t = threadIdx.x;
    if (t < FF_) {
        // channel-major layout: coalesced across lanes for each channel
        const float* cp = output + ((size_t)b * (A_ * NCH_) + (size_t)a * NCH_) * FF_ + t;
        float rx = cp[0 * FF_], ry = cp[1 * FF_];
        float rw = cp[2 * FF_], rh = cp[3 * FF_], ro = cp[4 * FF_];
        const int j = t / F_, i = t % F_;

        float sx = sigmoidf(rx), sy = sigmoidf(ry);
        float wa = c_anch[6 + a][0], ha = c_anch[6 + a][1];
        float px = sx + (float)i, py = sy + (float)j;
        float pw = __expf(rw) * wa, ph = __expf(rh) * ha;
        float parea = pw * ph;

        // ignore-mask IoU vs all truths + match scan (last writer wins,
        // matching scatter-with-duplicates semantics)
        float bestIou = 0.f; int win = -1;
        float af = (float)a, fi = (float)i, fj = (float)j;
        for (int m = 0; m < M_; ++m) {
            const float* T = &sT[m * TRUTH_STRIDE];
            float tbx = T[0], tby = T[1], tbw = T[2], tbh = T[3], val = T[4];
            float tlx = fmaxf(px - pw * 0.5f, tbx - tbw * 0.5f);
            float tly = fmaxf(py - ph * 0.5f, tby - tbh * 0.5f);
            float brx = fminf(px + pw * 0.5f, tbx + tbw * 0.5f);
            float bry = fminf(py + ph * 0.5f, tby + tbh * 0.5f);
            float en = ((tlx < brx) && (tly < bry)) ? 1.f : 0.f;
            float inter = (brx - tlx) * (bry - tly) * en;
            float iou = inter / (parea + tbw * tbh - inter);
            bestIou = fmaxf(bestIou, iou * val);
            if (T[5] > 0.5f && T[6] == af && T[7] == fi && T[8] == fj) win = m;
        }
        float ignoreMask = (bestIou <= 0.5f) ? 1.f : 0.f;
        float mk = (win >= 0) ? 1.f : 0.f;
        float twt = 0.f, tht = 0.f, s = 0.f; int clsIdx = -1;
        if (win >= 0) {
            const float* T = &sT[win * TRUTH_STRIDE];
            twt = T[9]; tht = T[10]; s = T[11]; clsIdx = (int)T[12];
        }
        float objMask = (win >= 0) ? 1.f : ignoreMask;
        float s2 = s * s;   // tgt_scale^2 (s==0 when unmatched)

        // xy: target is exactly 0 (txf=tyf=0), weight = tgt_scale^2
        {
            float p = sx * mk;
            lxy += bce_term(p, 0.f, s2);  ll2 += p * p;
            p = sy * mk;
            lxy += bce_term(p, 0.f, s2);  ll2 += p * p;
        }
        // wh: 0.5 * sum((s*raw - s*twt)^2)
        {
            float dw = (rw - twt) * mk * s;
            float dh = (rh - tht) * mk * s;
            lwh += 0.5f * (dw * dw + dh * dh);
            ll2 += dw * dw + dh * dh;
        }
        // obj
        {
            float p  = sigmoidf(ro) * objMask;
            float tt = mk * objMask;
            lobj += bce_term(p, tt, 1.0f);
            float d = p - tt; ll2 += d * d;
        }
        // cls (streamed)
        for (int c = 0; c < NC_; ++c) {
            float p  = sigmoidf(cp[(5 + c) * FF_]) * mk;
            float tt = (c == clsIdx) ? mk : 0.f;
            lcls += bce_term(p, tt, 1.0f);
            float d = p - tt; ll2 += d * d;
        }
    }

    // block reduction: wave shuffles -> LDS -> thread 0
    float vals[5] = {lxy, lwh, lobj, lcls, ll2};
    int lane = threadIdx.x & 31, wid = threadIdx.x >> 5;
    #pragma unroll
    for (int c = 0; c < 5; ++c) {
        float v = vals[c];
        for (int off = 16; off > 0; off >>= 1) v += __shfl_down(v, off, 32);
        if (lane == 0) sRed[wid * 5 + c] = v;
    }
    __syncthreads();
    if (threadIdx.x == 0) {
        #pragma unroll
        for (int c = 0; c < 5; ++c) {
            float sum = 0.f;
            for (int w = 0; w < 12; ++w) sum += sRed[w * 5 + c];
            partials[c * NB_ + blockIdx.x] = sum;
        }
    }
}

// ------------------------------------------------------------------
// Kernel 3: one wave; WMMA f32 16x16x4 dot-with-ones reduction of the
// 384 per-block partials per component (full fp32, C accumulates).
// ------------------------------------------------------------------
__global__ __launch_bounds__(32) void yolo_final_kernel(
    const float* __restrict__ partials, float* __restrict__ out) {
    const int lane = threadIdx.x;
    float res[5];
#if __has_builtin(__builtin_amdgcn_wmma_f32_16x16x4_f32)
    const v2f onesB = {1.0f, 1.0f};          // B = 4x16 all-ones
    const int m = lane & 15, kh = lane >> 4; // A 16x4 layout: lanes 0-15 K=0,1; 16-31 K=2,3
    #pragma unroll
    for (int c = 0; c < 5; ++c) {
        v8f acc = {};
        for (int chunk = 0; chunk < NB_ / 64; ++chunk) {
            int idx = c * NB_ + chunk * 64 + m * 4 + kh * 2;
            v2f av; av.x = partials[idx]; av.y = partials[idx + 1];
            // D[m][n] += sum_k A[m][k]  (row sums, replicated over n)
            acc = __builtin_amdgcn_wmma_f32_16x16x4_f32(
                false, av, false, onesB, (short)0, acc, false, false);
        }
        float ssum = acc[0] + acc[1] + acc[2] + acc[3] +
                     acc[4] + acc[5] + acc[6] + acc[7];   // M=0..7 (lanes<16) / 8..15
        ssum += __shfl_down(ssum, 16, 32);                // combine halves (lane 0 valid)
        res[c] = ssum;
    }
#else
    #pragma unroll
    for (int c = 0; c < 5; ++c) {
        float v = 0.f;
        for (int idx = lane; idx < NB_; idx += 32) v += partials[c * NB_ + idx];
        for (int off = 16; off > 0; off >>= 1) v += __shfl_down(v, off, 32);
        res[c] = v;
    }
#endif
    if (lane == 0) {
        out[0] = res[0] + res[1] + res[2] + res[3];  // total (l2 excluded, per reference)
        out[1] = res[0];  // loss_xy
        out[2] = res[1];  // loss_wh
        out[3] = res[2];  // loss_obj
        out[4] = res[3];  // loss_cls
        out[5] = res[4];  // loss_l2
    }
}

// ------------------------------------------------------------------
extern "C" void kernel_launch(void* const* d_in, const int* in_sizes, int n_in,
                              void* d_out, int out_size, void* d_ws, size_t ws_size,
                              hipStream_t stream) {
    (void)in_sizes; (void)n_in; (void)out_size; (void)ws_size;
    const float* output = (const float*)d_in[0];
    const float* labels = (const float*)d_in[1];
    float* out      = (float*)d_out;
    float* truth    = (float*)d_ws;               // B*M*16 floats
    float* partials = truth + TRUTH_FLOATS;       // 5*NB_ floats

    yolo_truth_kernel<<<(B_ * M_ + 255) / 256, 256, 0, stream>>>(labels, truth);
    yolo_cell_kernel<<<NB_, 384, 0, stream>>>(output, truth, partials);
    yolo_final_kernel<<<1, 32, 0, stream>>>(partials, out);
}